// NearestNeighbor_53025666236461
// MI455X (gfx1250) — compile-verified
//
#include <hip/hip_runtime.h>

typedef __attribute__((ext_vector_type(2))) float v2f;
typedef __attribute__((ext_vector_type(8))) float v8f;

#define HB 96
#define WB 96
#define LB (HB*WB)        // 9216 bank entries
#define CCH 4
#define KW 3
#define DD (CCH*KW*KW)    // 36 patch dim
#define NPAIR (DD/2)      // 18 K-pairs
#define HH 64
#define WW 64
#define LQ (HH*WW)        // 4096 queries per batch
#define BB 4
#define TT 4
#define NQ (BB*LQ)        // 16384 queries
#define NSTEP (TT-1)

#define WPB 8                        // waves per block (256 threads)
#define QPW 32                       // queries per wave: two 16-row M tiles
#define QPB (WPB*QPW)                // 256 queries per block
#define BLOCKS_PER_STEP (NQ/QPB)     // 64

#define CH_ENT 128                   // bank entries per staged LDS chunk
#define CH_TILES (CH_ENT/16)         // 8 N-tiles per chunk
#define NCHUNK (LB/CH_ENT)           // 72 chunks
#define STG_PER_THR (NPAIR*CH_ENT/256) // 9 v2f per thread per chunk

// Bank in K-pair-interleaved layout: bankP[(d>>1)*LB*2 + m*2 + (d&1)] = bank[m][d]
// -> one b64 per WMMA B fragment. bank_sq[m] = sum_d bank[m][d]^2.
__global__ void nn_prep_bank(const float* __restrict__ data,
                             float* __restrict__ bankP,
                             float* __restrict__ bank_sq) {
    int m = blockIdx.x * blockDim.x + threadIdx.x;
    if (m >= LB) return;
    int mi = m / WB, mj = m % WB;
    float sq = 0.f;
    for (int c = 0; c < CCH; ++c)
        for (int di = 0; di < KW; ++di)
            for (int dj = 0; dj < KW; ++dj) {
                int ii = mi + di - 1; ii += (ii < 0) ? HB : 0; ii -= (ii >= HB) ? HB : 0;
                int jj = mj + dj - 1; jj += (jj < 0) ? WB : 0; jj -= (jj >= WB) ? WB : 0;
                float v = data[c*LB + ii*WB + jj];
                int d = c*9 + di*3 + dj;
                bankP[(d >> 1)*(LB*2) + m*2 + (d & 1)] = v;
                sq += v*v;
            }
    bank_sq[m] = sq;
}

// all_steps[:,0] = x[:,0]
__global__ void nn_copy_t0(const float* __restrict__ x, float* __restrict__ out) {
    int i = blockIdx.x * blockDim.x + threadIdx.x;  // over B*C*H*W
    if (i >= BB*CCH*LQ) return;
    int b = i / (CCH*LQ); int rest = i - b*(CCH*LQ);
    out[(b*TT)*(CCH*LQ) + rest] = x[(b*TT)*(CCH*LQ) + rest];
}

// Each wave owns 32 queries (two WMMA M-tiles sharing every B fragment).
// Bank is streamed through double-buffered LDS chunks shared by all 8 waves:
//   iter c: [ds_store chunk c+1 from regs][issue global loads chunk c+2]
//           [consume chunk c: 8 tiles x (18 WMMA + argmin)] [barrier]
__global__ void __launch_bounds__(256)
nn_step(const float* __restrict__ state,   // full [B,T,C,H,W]; read slice cur_t
        int cur_t,
        const float* __restrict__ x,       // target = x[:, step+1]
        int step,
        const float* __restrict__ data,    // center gather source [C,HB,WB]
        const float* __restrict__ bankP,
        const float* __restrict__ bank_sq,
        float* __restrict__ out,           // full [B,T,C,H,W]; write slice step+1
        float* __restrict__ partials) {
    __shared__ v2f   s_bank[2][NPAIR*CH_ENT];   // 2 x 18KB
    __shared__ float s_bsq[2][CH_ENT];
    __shared__ int   s_idx[WPB*QPW];
    __shared__ float s_loss[WPB];

    const int tid   = threadIdx.x;
    const int wave  = tid >> 5;
    const int lane  = tid & 31;
    const int lhalf = lane >> 4;     // K-pair selector within each K4 chunk
    const int l16   = lane & 15;

    const int qbase = (blockIdx.x*WPB + wave)*QPW;

    // ---- A fragments for two M tiles (rows qbase+l16 and qbase+16+l16) ----
    v2f a0[9], a1[9];
#pragma unroll
    for (int mt = 0; mt < 2; ++mt) {
        const int q  = qbase + mt*16 + l16;
        const int b  = q >> 12;
        const int l  = q & (LQ - 1);
        const int qi = l >> 6;
        const int qj = l & (WW - 1);
#pragma unroll
        for (int kk = 0; kk < 9; ++kk) {
#pragma unroll
            for (int h = 0; h < 2; ++h) {
                int d  = kk*4 + 2*lhalf + h;
                int c  = d / 9, rem = d - c*9;
                int di = rem / 3, dj = rem - di*3;
                int ii = (qi + di - 1) & (HH - 1);
                int jj = (qj + dj - 1) & (WW - 1);
                float v = state[((b*TT + cur_t)*CCH + c)*LQ + ii*WW + jj];
                if (mt == 0) a0[kk][h] = v; else a1[kk][h] = v;
            }
        }
    }

    // ---- staging pipeline state ----
    v2f   stg[STG_PER_THR];
    float stg_sq = 0.f;
    const bool has_sq = (tid < CH_ENT);

    auto issue_loads = [&](int c) {
        const int m0 = c*CH_ENT;
#pragma unroll
        for (int t = 0; t < STG_PER_THR; ++t) {
            int idx = t*256 + tid;          // = p*CH_ENT + e
            int p = idx >> 7;
            int e = idx & (CH_ENT - 1);
            stg[t] = *(const v2f*)(bankP + (p*LB + m0 + e)*2);
        }
        if (has_sq) stg_sq = bank_sq[m0 + tid];
    };
    auto store_stage = [&](int buf) {
#pragma unroll
        for (int t = 0; t < STG_PER_THR; ++t)
            s_bank[buf][t*256 + tid] = stg[t];
        if (has_sq) s_bsq[buf][tid] = stg_sq;
    };

    float best_val0[8], best_val1[8];
    int   best_idx0[8], best_idx1[8];
#pragma unroll
    for (int r = 0; r < 8; ++r) {
        best_val0[r] = 3.0e38f; best_idx0[r] = 0;
        best_val1[r] = 3.0e38f; best_idx1[r] = 0;
    }

    issue_loads(0);
    store_stage(0);
    issue_loads(1);
    __syncthreads();

    for (int c = 0; c < NCHUNK; ++c) {
        const int buf = c & 1;
        if (c + 1 < NCHUNK) store_stage(buf ^ 1);   // chunk c+1 (loads 1 iter old)
        if (c + 2 < NCHUNK) issue_loads(c + 2);     // overlap with consume below

#pragma unroll
        for (int t = 0; t < CH_TILES; ++t) {
            const int   eo  = t*16 + l16;           // entry within chunk (N = l16)
            const float bsq = s_bsq[buf][eo];
            const int   m   = c*CH_ENT + eo;        // global bank index
            v2f bf[9];
#pragma unroll
            for (int kk = 0; kk < 9; ++kk)
                bf[kk] = s_bank[buf][(kk*2 + lhalf)*CH_ENT + eo];

            v8f acc0 = {0.f,0.f,0.f,0.f,0.f,0.f,0.f,0.f};
            v8f acc1 = {0.f,0.f,0.f,0.f,0.f,0.f,0.f,0.f};
#pragma unroll
            for (int kk = 0; kk < 9; ++kk) {
                acc0 = __builtin_amdgcn_wmma_f32_16x16x4_f32(
                    false, a0[kk], false, bf[kk], (short)0, acc0, false, false);
                acc1 = __builtin_amdgcn_wmma_f32_16x16x4_f32(
                    false, a1[kk], false, bf[kk], (short)0, acc1, false, false);
            }
            // D layout: VGPR r of this lane = (M = r + 8*lhalf, N = l16)
#pragma unroll
            for (int r = 0; r < 8; ++r) {
                float c0 = fmaf(-2.f, acc0[r], bsq);   // d2 - q_sq (argmin-equiv)
                if (c0 < best_val0[r]) { best_val0[r] = c0; best_idx0[r] = m; }
                float c1 = fmaf(-2.f, acc1[r], bsq);
                if (c1 < best_val1[r]) { best_val1[r] = c1; best_idx1[r] = m; }
            }
        }
        __syncthreads();
    }

    // argmin across the 16 lanes of each half; ties -> smaller bank index
#pragma unroll
    for (int r = 0; r < 8; ++r) {
#pragma unroll
        for (int off = 1; off <= 8; off <<= 1) {
            float ov0 = __shfl_xor(best_val0[r], off, 32);
            int   oi0 = __shfl_xor(best_idx0[r], off, 32);
            if (ov0 < best_val0[r] || (ov0 == best_val0[r] && oi0 < best_idx0[r])) {
                best_val0[r] = ov0; best_idx0[r] = oi0;
            }
            float ov1 = __shfl_xor(best_val1[r], off, 32);
            int   oi1 = __shfl_xor(best_idx1[r], off, 32);
            if (ov1 < best_val1[r] || (ov1 == best_val1[r] && oi1 < best_idx1[r])) {
                best_val1[r] = ov1; best_idx1[r] = oi1;
            }
        }
    }
    if (l16 == 0) {
#pragma unroll
        for (int r = 0; r < 8; ++r) {
            s_idx[wave*QPW +      lhalf*8 + r] = best_idx0[r];
            s_idx[wave*QPW + 16 + lhalf*8 + r] = best_idx1[r];
        }
    }
    __syncthreads();

    // each lane owns one of the wave's 32 queries: gather center, write, loss
    const int qidx = qbase + lane;
    const int b    = qidx >> 12;
    const int l    = qidx & (LQ - 1);
    const int best = s_idx[wave*QPW + lane];
    const int t_out = step + 1;
    float lsum = 0.f;
#pragma unroll
    for (int c = 0; c < CCH; ++c) {
        float v   = data[c*LB + best];     // center of patch == source pixel
        int   oix = ((b*TT + t_out)*CCH + c)*LQ + l;
        out[oix] = v;
        float dv = v - x[((b*TT + t_out)*CCH + c)*LQ + l];
        lsum += dv*dv;
    }
#pragma unroll
    for (int off = 1; off < 32; off <<= 1)
        lsum += __shfl_xor(lsum, off, 32);
    if (lane == 0) s_loss[wave] = lsum;
    __syncthreads();
    if (tid == 0) {
        float s = 0.f;
        for (int w = 0; w < WPB; ++w) s += s_loss[w];
        partials[step*BLOCKS_PER_STEP + blockIdx.x] = s;
    }
}

// Deterministic sequential reduction of the 3*64 block partials.
__global__ void nn_finish_loss(const float* __restrict__ partials,
                               float* __restrict__ out_loss) {
    if (threadIdx.x == 0 && blockIdx.x == 0) {
        float s = 0.f;
        for (int i = 0; i < NSTEP*BLOCKS_PER_STEP; ++i) s += partials[i];
        out_loss[0] = s / (float)(NSTEP * BB * CCH * LQ);
    }
}

extern "C" void kernel_launch(void* const* d_in, const int* in_sizes, int n_in,
                              void* d_out, int out_size, void* d_ws, size_t ws_size,
                              hipStream_t stream) {
    const float* x    = (const float*)d_in[0];   // [B,T,C,H,W] fp32
    const float* data = (const float*)d_in[1];   // [1,C,96,96] fp32
    float* out = (float*)d_out;                  // [B,T,C,H,W] steps + [1] loss
    float* ws  = (float*)d_ws;

    float* bankP    = ws;                 // 36*9216 floats (pair-interleaved)
    float* bank_sq  = bankP + DD*LB;      // 9216 floats
    float* partials = bank_sq + LB;       // 192 floats

    nn_prep_bank<<<(LB + 255)/256, 256, 0, stream>>>(data, bankP, bank_sq);
    nn_copy_t0<<<(BB*CCH*LQ + 255)/256, 256, 0, stream>>>(x, out);
    for (int s = 0; s < NSTEP; ++s) {
        // step s reads out slice t=s (slice 0 copied from x[:,0]),
        // writes out slice t=s+1; stream order provides the dependency
        nn_step<<<BLOCKS_PER_STEP, 256, 0, stream>>>(
            out, s, x, s, data, bankP, bank_sq, out, partials);
    }
    nn_finish_loss<<<1, 32, 0, stream>>>(partials, out + BB*TT*CCH*LQ);
}